// EdgeNet_22892175688228
// MI455X (gfx1250) — compile-verified
//
#include <hip/hip_runtime.h>

typedef __attribute__((ext_vector_type(2))) float v2f;
typedef __attribute__((ext_vector_type(8))) float v8f;

#define IN_F 256
#define H_F  128
#define OUT_F 64

// ---------------- degree / normalization ----------------

__global__ void k_deg_init(float* deg, int n) {
    int i = blockIdx.x * blockDim.x + threadIdx.x;
    if (i < n) deg[i] = 1.0f;   // self-loop contributes 1
}

__global__ void k_deg_accum(const int* __restrict__ dst, float* deg, int nE) {
    int e = blockIdx.x * blockDim.x + threadIdx.x;
    if (e < nE) atomicAdd(&deg[dst[e]], 1.0f);
}

__global__ void k_rsqrt_inplace(float* d, int n) {
    int i = blockIdx.x * blockDim.x + threadIdx.x;
    if (i < n) d[i] = rsqrtf(d[i]);   // deg >= 1 always (self-loops)
}

// ---------------- fp32 WMMA GEMM:  Y[nrows x Ko] = op(X)[nrows x Ki] @ W[Ki x Ko] ----------------
// One wave computes 16 rows x Ko cols. A frag reused across all column tiles.
// V_WMMA_F32_16X16X4_F32 layouts (wave32):
//   A (16x4): lane l -> row M=l%16; VGPR0/1 = K = 2*(l/16) + {0,1}
//   B (4x16): lane l -> col N=l%16; VGPR0/1 = K = 2*(l/16) + {0,1}
//   C/D     : VGPR r, lane l -> M = r + 8*(l/16), N = l%16

template <int Ki, int Ko, bool RELU_A>
__global__ __launch_bounds__(256)
void k_gemm_wmma(const float* __restrict__ X, const float* __restrict__ W,
                 float* __restrict__ Y, int nrows) {
    constexpr int CT = Ko / 16;              // column tiles
    const int wave = threadIdx.x >> 5;
    const int lane = threadIdx.x & 31;
    const int rowTile = blockIdx.x * 8 + wave;   // 8 waves per block
    const int row0 = rowTile * 16;
    if (row0 >= nrows) return;               // wave-uniform: EXEC stays all-ones

    const int m    = lane & 15;
    const int half = lane >> 4;              // 0 or 1
    const float* xrow = X + (size_t)(row0 + m) * Ki + 2 * half;

    v8f acc[CT] = {};

    for (int kb = 0; kb < Ki; kb += 4) {
        // prefetch ahead in this row (gfx1250 global_prefetch_b8)
        if (kb + 32 < Ki) __builtin_prefetch(xrow + kb + 32, 0, 3);

        float ax = xrow[kb];
        float ay = xrow[kb + 1];
        if (RELU_A) { ax = fmaxf(ax, 0.0f); ay = fmaxf(ay, 0.0f); }
        v2f a = { ax, ay };

        const int krow = kb + 2 * half;
        const float* w0 = W + (size_t)krow * Ko;   // B row krow
        const float* w1 = w0 + Ko;                 // B row krow+1
#pragma unroll
        for (int j = 0; j < CT; ++j) {
            v2f b = { w0[j * 16 + m], w1[j * 16 + m] };
            acc[j] = __builtin_amdgcn_wmma_f32_16x16x4_f32(
                false, a, false, b, (short)0, acc[j], false, false);
        }
    }

#pragma unroll
    for (int j = 0; j < CT; ++j) {
#pragma unroll
        for (int r = 0; r < 8; ++r) {
            const int M = r + 8 * half;
            Y[(size_t)(row0 + M) * Ko + j * 16 + m] = acc[j][r];
        }
    }
}

// ---------------- aggregation ----------------
// init: agg = lin * dinv^2 (self-loop, norm = dinv[i]*dinv[i]) + bias
__global__ void k_agg_init(const float* __restrict__ lin, const float* __restrict__ dinv,
                           const float* __restrict__ bias, float* __restrict__ agg,
                           int n, int F) {
    int t = blockIdx.x * blockDim.x + threadIdx.x;
    int i = t / F;
    int f = t - i * F;
    if (i >= n) return;
    float w = dinv[i];
    agg[t] = lin[t] * (w * w) + bias[f];
}

// edges: agg[dst] += lin[src] * dinv[src]*dinv[dst], float4 chunks via atomics
__global__ void k_agg_edges(const float* __restrict__ lin, const int* __restrict__ src,
                            const int* __restrict__ dst, const float* __restrict__ dinv,
                            float* agg, int nE, int F4) {
    int t = blockIdx.x * blockDim.x + threadIdx.x;
    int e = t / F4;
    int c = t - e * F4;
    if (e >= nE) return;
    int s = src[e], d = dst[e];
    float w = dinv[s] * dinv[d];
    const float4 v = ((const float4*)(lin + (size_t)s * F4 * 4))[c];
    float* o = agg + (size_t)d * F4 * 4 + (size_t)c * 4;
    atomicAdd(o + 0, v.x * w);
    atomicAdd(o + 1, v.y * w);
    atomicAdd(o + 2, v.z * w);
    atomicAdd(o + 3, v.w * w);
}

// ---------------- decoder: wave-per-edge dot over 64 dims ----------------
__global__ void k_decode(const float* __restrict__ z,
                         const int* __restrict__ pos, const int* __restrict__ neg,
                         float* __restrict__ out, int EP) {
    int wid  = (blockIdx.x * blockDim.x + threadIdx.x) >> 5;
    int lane = threadIdx.x & 31;
    if (wid >= 2 * EP) return;   // wave-uniform
    int u, v;
    if (wid < EP) { u = pos[wid];       v = pos[EP + wid]; }
    else          { u = neg[wid - EP];  v = neg[EP + (wid - EP)]; }
    const float* zu = z + (size_t)u * OUT_F;
    const float* zv = z + (size_t)v * OUT_F;
    float s = zu[lane] * zv[lane] + zu[lane + 32] * zv[lane + 32];
#pragma unroll
    for (int off = 16; off > 0; off >>= 1)
        s += __shfl_down(s, off, 32);
    if (lane == 0) out[wid] = s;
}

// ---------------- launch ----------------

extern "C" void kernel_launch(void* const* d_in, const int* in_sizes, int n_in,
                              void* d_out, int out_size, void* d_ws, size_t ws_size,
                              hipStream_t stream) {
    const float* x    = (const float*)d_in[0];
    const int*   ei   = (const int*)  d_in[1];   // [2, E] row-major
    const int*   pei  = (const int*)  d_in[2];   // [2, EP]
    const int*   nei  = (const int*)  d_in[3];
    const float* W1   = (const float*)d_in[4];
    const float* b1   = (const float*)d_in[5];
    const float* W2   = (const float*)d_in[6];
    const float* b2   = (const float*)d_in[7];
    float* out = (float*)d_out;

    const int N  = in_sizes[0] / IN_F;
    const int E  = in_sizes[1] / 2;
    const int EP = in_sizes[2] / 2;
    const int* src = ei;
    const int* dst = ei + E;

    // workspace layout (floats)
    float* dinv = (float*)d_ws;
    size_t npad = ((size_t)N + 255) & ~(size_t)255;
    float* bufA = dinv + npad;                 // hlin, later reused as zlin
    float* h    = bufA + (size_t)N * H_F;      // layer-1 aggregated output
    float* z    = h    + (size_t)N * H_F;      // layer-2 aggregated output

    const int B = 256;
    // 1) degree + rsqrt
    k_deg_init<<<(N + B - 1) / B, B, 0, stream>>>(dinv, N);
    k_deg_accum<<<(E + B - 1) / B, B, 0, stream>>>(dst, dinv, E);
    k_rsqrt_inplace<<<(N + B - 1) / B, B, 0, stream>>>(dinv, N);

    // 2) layer 1: hlin = x @ W1   (fp32 WMMA)
    {
        int rowTiles = (N + 15) / 16;
        int blocks   = (rowTiles + 7) / 8;
        k_gemm_wmma<IN_F, H_F, false><<<blocks, B, 0, stream>>>(x, W1, bufA, N);
    }
    // 3) aggregate layer 1: h = scatter(norm * hlin) + b1   (relu fused into GEMM2)
    {
        int tot = N * H_F;
        k_agg_init<<<(tot + B - 1) / B, B, 0, stream>>>(bufA, dinv, b1, h, N, H_F);
        long long te = (long long)E * (H_F / 4);
        k_agg_edges<<<(int)((te + B - 1) / B), B, 0, stream>>>(bufA, src, dst, dinv, h, E, H_F / 4);
    }
    // 4) layer 2: zlin = relu(h) @ W2   (fp32 WMMA, relu fused on A-load)
    {
        int rowTiles = (N + 15) / 16;
        int blocks   = (rowTiles + 7) / 8;
        k_gemm_wmma<H_F, OUT_F, true><<<blocks, B, 0, stream>>>(h, W2, bufA, N);
    }
    // 5) aggregate layer 2: z = scatter(norm * zlin) + b2
    {
        int tot = N * OUT_F;
        k_agg_init<<<(tot + B - 1) / B, B, 0, stream>>>(bufA, dinv, b2, z, N, OUT_F);
        long long te = (long long)E * (OUT_F / 4);
        k_agg_edges<<<(int)((te + B - 1) / B), B, 0, stream>>>(bufA, src, dst, dinv, z, E, OUT_F / 4);
    }
    // 6) decode: out[i] = <z_u, z_v> for pos then neg edges (wave per edge)
    {
        long long waves = 2LL * EP;
        long long thr   = waves * 32;
        k_decode<<<(int)((thr + B - 1) / B), B, 0, stream>>>(z, pei, nei, out, EP);
    }
}